// AAMsoftmax_3582002724978
// MI455X (gfx1250) — compile-verified
//
#include <hip/hip_runtime.h>
#include <hip/hip_bf16.h>

typedef float v2f __attribute__((ext_vector_type(2)));
typedef float v4f __attribute__((ext_vector_type(4)));
typedef float v8f __attribute__((ext_vector_type(8)));

// AAM-softmax constants (m=0.2, s=30)
#define K_COSM 0.98006657784124163f   // cos(0.2)
#define K_SINM 0.19866933079506122f   // sin(0.2)
#define K_TH  (-0.98006657784124163f) // cos(pi-0.2)
#define K_MM  0.039733866159012244f   // sin(pi-0.2)*0.2
#define K_S   30.0f
#define NEGBIG (-1.0e30f)

struct __align__(16) Partial {
    float mx;   // max logit in this chunk
    float sum;  // sum exp(logit - mx)
    float lab;  // logit at label column (NEGBIG if not in chunk)
    int   idx;  // argmax column
};

// Swizzled row layout (192 floats): element k -> ((k>>1)&1)*96 + (k>>2)*2 + (k&1)
// => each WMMA lane's K-stream (its ksel half) is contiguous, enabling b128 loads:
//    lane offset for K-step s (K = 4s..4s+3) is ksel*96 + 2*s.
__device__ __forceinline__ int swz192(int k) {
    return ((k >> 1) & 1) * 96 + ((k >> 2) << 1) + (k & 1);
}

// ---------------------------------------------------------------------------
// Row L2 normalization (swizzled output): one wave per row of 192 (6/lane)
// ---------------------------------------------------------------------------
__global__ __launch_bounds__(256) void l2norm_rows_swz(const float* __restrict__ in,
                                                       float* __restrict__ out,
                                                       int nrows) {
    const int wave = threadIdx.x >> 5;
    const int lane = threadIdx.x & 31;
    const int row  = blockIdx.x * 8 + wave;
    if (row >= nrows) return;                 // wave-uniform exit
    const float* p = in + (size_t)row * 192;
    float v[6];
    float ss = 0.0f;
#pragma unroll
    for (int j = 0; j < 6; ++j) {
        v[j] = p[lane + 32 * j];
        ss += v[j] * v[j];
    }
#pragma unroll
    for (int msk = 16; msk >= 1; msk >>= 1)
        ss += __shfl_xor(ss, msk, 32);
    const float scale = 1.0f / fmaxf(sqrtf(ss), 1e-12f);
    float* q = out + (size_t)row * 192;
#pragma unroll
    for (int j = 0; j < 6; ++j)
        q[swz192(lane + 32 * j)] = v[j] * scale;
}

// ---------------------------------------------------------------------------
// Fused GEMM (xn @ wn^T) + margin + online split-softmax partials.
// Block: 256 threads (8 waves). Block tile: 32 rows x 512 cols.
// Wave tile: 32 rows x 64 cols = 2x4 v8f accumulators, V_WMMA_F32_16X16X4_F32.
// Branch-free K loop (EXEC all-ones for WMMA); b128 operand loads.
// ---------------------------------------------------------------------------
__global__ __launch_bounds__(256) void aam_gemm_partial(
    const float* __restrict__ xn, const float* __restrict__ wn,
    const long long* __restrict__ label, Partial* __restrict__ parts,
    int C, int nchunks) {
    const int chunk   = blockIdx.x;
    const int rowbase = blockIdx.y * 32;
    const int wave    = threadIdx.x >> 5;
    const int lane    = threadIdx.x & 31;
    const int n       = lane & 15;   // N (B/C/D frag) or M (A frag) index
    const int ksel    = lane >> 4;   // K-pair select (A/B) / M-half (C/D)
    const int wcol    = chunk * 512 + wave * 64;

    __shared__ int     labs[32];
    __shared__ Partial st[8][32];
    if (threadIdx.x < 32) labs[threadIdx.x] = (int)label[rowbase + threadIdx.x];
    __syncthreads();

    bool valid[4];
#pragma unroll
    for (int t = 0; t < 4; ++t) valid[t] = (wcol + t * 16) < C;  // C % 16 == 0

    // clamp invalid tiles to col 0 (in-bounds garbage, masked in epilogue)
    const float* arow0 = xn + (size_t)(rowbase + n) * 192 + ksel * 96;
    const float* arow1 = arow0 + (size_t)16 * 192;
    const float* brow[4];
#pragma unroll
    for (int t = 0; t < 4; ++t)
        brow[t] = wn + (size_t)(valid[t] ? (wcol + t * 16 + n) : n) * 192 + ksel * 96;

    v8f acc[2][4];
#pragma unroll
    for (int rt = 0; rt < 2; ++rt)
#pragma unroll
        for (int t = 0; t < 4; ++t)
#pragma unroll
            for (int r = 0; r < 8; ++r) acc[rt][t][r] = 0.0f;

    // 24 iterations x 2 K-steps (K=8 per iter), 16 WMMAs per iteration
#pragma unroll 2
    for (int i = 0; i < 24; ++i) {
        v4f a0 = *(const v4f*)(arow0 + 4 * i);
        v4f a1 = *(const v4f*)(arow1 + 4 * i);
        v4f b[4];
#pragma unroll
        for (int t = 0; t < 4; ++t) b[t] = *(const v4f*)(brow[t] + 4 * i);
#pragma unroll
        for (int t = 0; t < 4; ++t) {
            v2f blo = __builtin_shufflevector(b[t], b[t], 0, 1);
            acc[0][t] = __builtin_amdgcn_wmma_f32_16x16x4_f32(
                false, __builtin_shufflevector(a0, a0, 0, 1), false, blo,
                (short)0, acc[0][t], false, false);
            acc[1][t] = __builtin_amdgcn_wmma_f32_16x16x4_f32(
                false, __builtin_shufflevector(a1, a1, 0, 1), false, blo,
                (short)0, acc[1][t], false, false);
        }
#pragma unroll
        for (int t = 0; t < 4; ++t) {
            v2f bhi = __builtin_shufflevector(b[t], b[t], 2, 3);
            acc[0][t] = __builtin_amdgcn_wmma_f32_16x16x4_f32(
                false, __builtin_shufflevector(a0, a0, 2, 3), false, bhi,
                (short)0, acc[0][t], false, false);
            acc[1][t] = __builtin_amdgcn_wmma_f32_16x16x4_f32(
                false, __builtin_shufflevector(a1, a1, 2, 3), false, bhi,
                (short)0, acc[1][t], false, false);
        }
    }

    // Epilogue (two-phase): per row slice, exp-free max/argmax/label butterflies,
    // then a single __expf per element and a pure-add butterfly for the sum.
    // C/D layout: VGPR r, lanes 0-15 -> M=r, lanes 16-31 -> M=r+8, lane%16 = N.
#pragma unroll
    for (int rt = 0; rt < 2; ++rt) {
#pragma unroll
        for (int r = 0; r < 8; ++r) {
            const int m  = rt * 16 + r + 8 * ksel;  // row within block tile
            const int gl = labs[m];
            float l[4];
            float mx = NEGBIG, lab = NEGBIG;
            int   idx = -1;
#pragma unroll
            for (int t = 0; t < 4; ++t) {
                float c    = acc[rt][t][r];
                float s2   = fminf(fmaxf(1.0f - c * c, 0.0f), 1.0f);
                float phi  = c * K_COSM - sqrtf(s2) * K_SINM;
                phi        = (c - K_TH > 0.0f) ? phi : (c - K_MM);
                const int col  = wcol + t * 16 + n;
                const bool isl = (col == gl);
                float lv = (isl ? phi : c) * K_S;
                lv = valid[t] ? lv : NEGBIG;      // select, no branch
                if (isl && valid[t]) lab = lv;
                l[t] = lv;
                if (lv > mx) { mx = lv; idx = col; }
            }
            // phase 1: exp-free butterflies for max/argmax/label over 16 lanes
#pragma unroll
            for (int msk = 1; msk < 16; msk <<= 1) {
                float omx  = __shfl_xor(mx, msk, 16);
                int   oidx = __shfl_xor(idx, msk, 16);
                lab = fmaxf(lab, __shfl_xor(lab, msk, 16));
                if (omx > mx) { mx = omx; idx = oidx; }
            }
            // phase 2: sum exp(l - rowmax); add-only butterfly
            float sum = 0.0f;
#pragma unroll
            for (int t = 0; t < 4; ++t) sum += __expf(l[t] - mx);
#pragma unroll
            for (int msk = 1; msk < 16; msk <<= 1)
                sum += __shfl_xor(sum, msk, 16);
            if (n == 0) {
                st[wave][m].mx  = mx;
                st[wave][m].sum = sum;
                st[wave][m].lab = lab;
                st[wave][m].idx = idx;
            }
        }
    }
    __syncthreads();

    // cross-wave merge (8 waves -> one partial per row) and store.
    // A fully-invalid wave has mx==NEGBIG; its garbage sum is annihilated by
    // the exp(p.mx - nm)==0 factor (every chunk has >=1 valid wave).
    if (threadIdx.x < 32) {
        const int m = threadIdx.x;
        float mx = NEGBIG, sum = 0.0f, lab = NEGBIG;
        int   idx = -1;
#pragma unroll
        for (int w = 0; w < 8; ++w) {
            Partial p = st[w][m];
            float nm = fmaxf(mx, p.mx);
            sum = sum * __expf(mx - nm) + p.sum * __expf(p.mx - nm);
            if (p.mx > mx) idx = p.idx;
            mx  = nm;
            lab = fmaxf(lab, p.lab);
        }
        Partial o;
        o.mx = mx; o.sum = sum; o.lab = lab; o.idx = idx;
        parts[(size_t)(rowbase + m) * nchunks + chunk] = o;
    }
}

// ---------------------------------------------------------------------------
// Merge chunk partials per row -> per-row CE loss + correctness -> reduce.
// ---------------------------------------------------------------------------
__global__ __launch_bounds__(1024) void aam_finalize(
    const Partial* __restrict__ parts, const long long* __restrict__ label,
    float* __restrict__ out, int nchunks, int B) {
    const int g = threadIdx.x;
    float loss = 0.0f, corr = 0.0f;
    if (g < B) {
        float mx = NEGBIG, sum = 0.0f, lab = NEGBIG;
        int   idx = -1;
        for (int c = 0; c < nchunks; ++c) {
            Partial p = parts[(size_t)g * nchunks + c];
            float nm = fmaxf(mx, p.mx);
            sum = sum * expf(mx - nm) + p.sum * expf(p.mx - nm);
            if (p.mx > mx) idx = p.idx;
            mx  = nm;
            lab = fmaxf(lab, p.lab);
        }
        loss = -(lab - mx - logf(sum));
        corr = (idx == (int)label[g]) ? 1.0f : 0.0f;
    }
    __shared__ float sl[1024];
    __shared__ float sc[1024];
    sl[g] = loss;
    sc[g] = corr;
    __syncthreads();
    for (int s = 512; s > 0; s >>= 1) {
        if (g < s) {
            sl[g] += sl[g + s];
            sc[g] += sc[g + s];
        }
        __syncthreads();
    }
    if (g == 0) {
        out[0] = sl[0] / (float)B;
        out[1] = sc[0] * 100.0f / (float)B;
    }
}

// ---------------------------------------------------------------------------
extern "C" void kernel_launch(void* const* d_in, const int* in_sizes, int n_in,
                              void* d_out, int out_size, void* d_ws, size_t ws_size,
                              hipStream_t stream) {
    (void)n_in; (void)out_size; (void)ws_size;
    const float*     x     = (const float*)d_in[0];      // [B, 192]
    const float*     w     = (const float*)d_in[1];      // [C, 192]
    const long long* label = (const long long*)d_in[2];  // [B] int64

    const int D = 192;
    const int B = in_sizes[2];
    const int C = in_sizes[1] / D;
    const int nchunks = (C + 511) / 512;

    // workspace: [ wn : C*D f32 ][ xn : B*D f32 ][ partials : B*nchunks ]
    float*   wn    = (float*)d_ws;
    float*   xn    = wn + (size_t)C * D;
    Partial* parts = (Partial*)(xn + (size_t)B * D);

    l2norm_rows_swz<<<(B + 7) / 8, 256, 0, stream>>>(x, xn, B);
    l2norm_rows_swz<<<(C + 7) / 8, 256, 0, stream>>>(w, wn, C);

    dim3 grid(nchunks, B / 32);
    aam_gemm_partial<<<grid, 256, 0, stream>>>(xn, wn, label, parts, C, nchunks);

    aam_finalize<<<1, 1024, 0, stream>>>(parts, label, (float*)d_out, nchunks, B);
}